// FusedXSALAKERAttention_54743653154996
// MI455X (gfx1250) — compile-verified
//
#include <hip/hip_runtime.h>

// ---------------------------------------------------------------------------
// FusedXSALAKERAttention for MI455X (gfx1250, wave32).
// fp32 contractions on V_WMMA_F32_16X16X4_F32. Hot kernel (Kmat @ alpha,
// 9 passes over the 512 MB Gram matrix) is LDS-staged: K slab moved by the
// Tensor Data Mover (tensor_load_to_lds + s_wait_tensorcnt) with D# padding
// chosen so WMMA A-fragment ds_load_b64 reads are bank-conflict-free;
// alpha chunk staged cooperatively with b128 LDS stores. 8 waves/block,
// 4 C-tiles per wave (A-fragment reuse x4).
// Workspace requirement: ~641 MB (Kmat dominates at 512 MB).
// ---------------------------------------------------------------------------

typedef __attribute__((ext_vector_type(2))) float v2f;
typedef __attribute__((ext_vector_type(8))) float v8f;
typedef __attribute__((ext_vector_type(4))) unsigned int u32x4;
typedef __attribute__((ext_vector_type(4))) int i32x4;
typedef __attribute__((ext_vector_type(8))) int i32x8;

constexpr int BB    = 2;
constexpr int SS    = 2048;
constexpr int DM    = 1024;
constexpr int HH    = 16;
constexpr int DH    = 64;
constexpr int RK    = 16;
constexpr int ITERS = 8;

// kmat_gemm tiling
constexpr int MSLAB = 128;   // output rows per block
constexpr int KC    = 32;    // k-chunk staged per step
constexpr int KPAD  = 34;    // LDS row stride of K slab (32 + 2 pad dwords)
constexpr int APAD  = 72;    // LDS row stride of alpha chunk

// workspace offsets (in floats)
constexpr size_t QKV      = (size_t)BB * HH * SS * DH;   // 4,194,304
constexpr size_t OFF_Q    = 0;
constexpr size_t OFF_K    = QKV;
constexpr size_t OFF_V    = 2 * QKV;
constexpr size_t OFF_ALPHA= 3 * QKV;
constexpr size_t OFF_RES  = 4 * QKV;                     // residual / raw out
constexpr size_t OFF_ATTN = 5 * QKV;                     // [B,S,D] pre-w_o
constexpr size_t OFF_LR   = 6 * QKV;                     // H*S*RK
constexpr size_t OFF_LTR  = OFF_LR  + (size_t)HH * SS * RK;
constexpr size_t OFF_QSQ  = OFF_LTR + (size_t)BB * HH * RK * DH;
constexpr size_t OFF_KSQ  = OFF_QSQ + (size_t)BB * HH * SS;
constexpr size_t OFF_KMAT = OFF_KSQ + (size_t)BB * HH * SS;   // B*H*S*S

#if defined(__has_builtin)
#if __has_builtin(__builtin_amdgcn_tensor_load_to_lds) && \
    __has_builtin(__builtin_amdgcn_s_wait_tensorcnt)
#define USE_TDM 1
#endif
#endif
#ifndef USE_TDM
#define USE_TDM 0
#endif

__device__ __forceinline__ v8f wmma4(v2f a, v2f b, v8f c) {
  // (neg_a, A, neg_b, B, c_mod, C, reuse_a, reuse_b)
  return __builtin_amdgcn_wmma_f32_16x16x4_f32(false, a, false, b, (short)0, c,
                                               false, false);
}

__device__ __forceinline__ float softplus_eps(float x) {
  return (x > 20.f ? x : log1pf(__expf(x))) + 1e-6f;
}

// --- 1) C[M,N] = X[M,K] @ W[N,K]^T  (M=B*S, N=K=1024) ----------------------
// mode 0: scatter to [B,H,S,Dh];  mode 1: direct [M,N] (final w_o GEMM)
__global__ __launch_bounds__(256) void proj_gemm_kernel(
    const float* __restrict__ X, const float* __restrict__ W,
    float* __restrict__ out, int mode) {
  const int lane = threadIdx.x & 31;
  const int lh   = lane >> 4;
  const int l16  = lane & 15;
  const int waveId = blockIdx.x * 8 + (threadIdx.x >> 5);
  const int tilesN = DM / 16;                       // 64
  const int tM = waveId / tilesN;
  const int tN = waveId % tilesN;
  const float* arow = X + (size_t)(tM * 16 + l16) * DM + 2 * lh;
  const float* brow = W + (size_t)(tN * 16 + l16) * DM + 2 * lh;
  v8f acc = {};
  for (int kk = 0; kk < DM; kk += 4) {
    float2 av = *(const float2*)(arow + kk);
    float2 bv = *(const float2*)(brow + kk);
    v2f a, b; a.x = av.x; a.y = av.y; b.x = bv.x; b.y = bv.y;
    acc = wmma4(a, b, acc);
  }
  const int n = tN * 16 + l16;
#pragma unroll
  for (int j = 0; j < 8; ++j) {
    const int m = tM * 16 + j + 8 * lh;
    if (mode == 0) {               // uniform branch (kernel arg)
      const int b_ = m / SS, s = m % SS, h = n / DH, d = n % DH;
      out[(((size_t)b_ * HH + h) * SS + s) * DH + d] = acc[j];
    } else {
      out[(size_t)m * DM + n] = acc[j];
    }
  }
}

// --- 2) per-row |q|^2, |k|^2 ----------------------------------------------
__global__ void sumsq_kernel(const float* __restrict__ q,
                             const float* __restrict__ k,
                             float* __restrict__ qsq, float* __restrict__ ksq) {
  const int i = blockIdx.x * blockDim.x + threadIdx.x;
  if (i >= BB * HH * SS) return;
  const float* qp = q + (size_t)i * DH;
  const float* kp = k + (size_t)i * DH;
  float sq = 0.f, sk = 0.f;
  for (int d = 0; d < DH; ++d) { sq += qp[d] * qp[d]; sk += kp[d] * kp[d]; }
  qsq[i] = sq; ksq[i] = sk;
}

// --- 3) Kmat[bh,s,t] = exp(-max(q2+k2-2 q.k,0)/(2bw^2)), diag masked -------
__global__ __launch_bounds__(256) void rbf_kernel_matrix(
    const float* __restrict__ q, const float* __restrict__ k,
    const float* __restrict__ qsq, const float* __restrict__ ksq,
    const float* __restrict__ bandwidth, float* __restrict__ Kmat) {
  const int lane = threadIdx.x & 31, lh = lane >> 4, l16 = lane & 15;
  const int waveId = blockIdx.x * 8 + (threadIdx.x >> 5);
  const int tiles = SS / 16;                        // 128
  const int bh  = waveId / (tiles * tiles);
  const int rem = waveId % (tiles * tiles);
  const int tS = rem / tiles, tT = rem % tiles;
  const float* qbase = q + ((size_t)bh * SS + tS * 16 + l16) * DH + 2 * lh;
  const float* kbase = k + ((size_t)bh * SS + tT * 16 + l16) * DH + 2 * lh;
  v8f acc = {};
  for (int dd = 0; dd < DH; dd += 4) {
    float2 av = *(const float2*)(qbase + dd);
    float2 bv = *(const float2*)(kbase + dd);
    v2f a, b; a.x = av.x; a.y = av.y; b.x = bv.x; b.y = bv.y;
    acc = wmma4(a, b, acc);
  }
  const float bw = softplus_eps(bandwidth[0]);
  const float invden = 1.f / (2.f * bw * bw);
  const float* qs  = qsq + (size_t)bh * SS;
  const int    t   = tT * 16 + l16;
  const float  kst = ksq[(size_t)bh * SS + t];
  float* ob = Kmat + (size_t)bh * SS * SS;
#pragma unroll
  for (int j = 0; j < 8; ++j) {
    const int s = tS * 16 + j + 8 * lh;
    float dist = fmaxf(qs[s] + kst - 2.f * acc[j], 0.f);
    ob[(size_t)s * SS + t] = (s == t) ? 0.f : __expf(-dist * invden);
  }
}

// --- 4) lr[h,s,r] = pos[s,:] @ head_proj[h,:,r] ----------------------------
__global__ void lr_kernel(const float* __restrict__ pos,
                          const float* __restrict__ hp,
                          float* __restrict__ lr) {
  const int i = blockIdx.x * blockDim.x + threadIdx.x;
  if (i >= HH * SS * RK) return;
  const int r = i % RK, s = (i / RK) % SS, h = i / (RK * SS);
  const float* pe  = pos + (size_t)s * RK;
  const float* hpp = hp + (size_t)h * RK * RK;
  float acc = 0.f;
  for (int rp = 0; rp < RK; ++rp) acc += pe[rp] * hpp[rp * RK + r];
  lr[i] = acc;
}

__global__ void zero_kernel(float* __restrict__ p, size_t n) {
  const size_t i = (size_t)blockIdx.x * blockDim.x + threadIdx.x;
  if (i < n) p[i] = 0.f;
}

// --- 5) HOT LOOP: resid = v - Kmat@alpha - lam*alpha   (mode 0) ------------
//                  out_raw = Kmat@alpha                 (mode 1)
// Block = 256 thr (8 waves) owns a 128x64 slab. K slab (128xKC) staged into
// LDS by the Tensor Data Mover each chunk (pad 2 dwords / 32 dwords -> LDS
// row stride KPAD=34 floats, bank-conflict-free A-fragment reads). Alpha
// chunk staged with b128 LDS stores (APAD=72: lane-half banks disjoint).
// Each wave keeps 4 accumulators and reuses its A fragment across 4 WMMAs.
__global__ __launch_bounds__(256) void kmat_gemm_kernel(
    const float* __restrict__ Kmat, const float* __restrict__ alpha,
    const float* __restrict__ v, const float* __restrict__ lambda_reg,
    float* __restrict__ outp, int mode) {
  __shared__ float ldsK[MSLAB * KPAD];   // 17,408 B
  __shared__ float ldsA[KC * APAD];      //  9,216 B
  const int tid  = threadIdx.x;
  const int lane = tid & 31, lh = lane >> 4, l16 = lane & 15;
  const int wv   = tid >> 5;                         // wave -> 16-row strip
  const int slabs = SS / MSLAB;                      // 16
  const int bh = blockIdx.x / slabs;
  const int ms = blockIdx.x % slabs;
  const float* Kg = Kmat + ((size_t)bh * SS + ms * MSLAB) * SS;
  const float* Ag = alpha + (size_t)bh * SS * DH;

  v8f acc[4] = {};

  for (int kk = 0; kk < SS; kk += KC) {
#if USE_TDM
    if (tid < 32) {   // wave 0 drives the TDM; EXEC uniform per wave
      const unsigned long long gaddr =
          (unsigned long long)(const void*)(Kg + kk);
      const unsigned int ldsb = (unsigned int)(unsigned long long)&ldsK[0];
      u32x4 g0;
      g0[0] = 1u;                                   // count=1 (valid user D#)
      g0[1] = ldsb;                                 // lds_addr
      g0[2] = (unsigned int)gaddr;                  // global_addr[31:0]
      g0[3] = ((unsigned int)(gaddr >> 32) & 0x1FFFFFFu) | (2u << 30); // type=2
      i32x8 g1;
      g1[0] = (int)((2u << 16) |                    // data_size = 4B
                    (1u << 20) |                    // pad_enable
                    (4u << 22) |                    // pad_interval: 32 dwords
                    (1u << 25));                    // pad_amount: 2 dwords
      g1[1] = (int)((unsigned)SS << 16);            // tensor_dim0[15:0]
      g1[2] = (int)((unsigned)SS << 16);            // dim0 hi=0 | tensor_dim1 lo
      g1[3] = (int)((unsigned)KC << 16);            // dim1 hi=0 | tile_dim0=KC
      g1[4] = (int)MSLAB;                           // tile_dim1=128, tile_dim2=0
      g1[5] = (int)SS;                              // tensor_dim0_stride[31:0]
      g1[6] = 0;
      g1[7] = 0;
      i32x4 gz = {0, 0, 0, 0};
#if __clang_major__ >= 23
      i32x8 gz8 = {0, 0, 0, 0, 0, 0, 0, 0};
      __builtin_amdgcn_tensor_load_to_lds(g0, g1, gz, gz, gz8, 0);
#else
      __builtin_amdgcn_tensor_load_to_lds(g0, g1, gz, gz, 0);
#endif
      __builtin_amdgcn_s_wait_tensorcnt(0);
    }
#else
    // fallback: cooperative copy, 2048 float2 -> 8 per thread
#pragma unroll
    for (int i = 0; i < 8; ++i) {
      const int idx = i * 256 + tid;
      const int r = idx >> 4, c2 = idx & 15;
      *(float2*)&ldsK[r * KPAD + c2 * 2] =
          *(const float2*)(Kg + (size_t)r * SS + kk + c2 * 2);
    }
#endif
    // stage alpha chunk: 32 x 64 floats = 512 float4 -> 2 per thread
#pragma unroll
    for (int i = 0; i < 2; ++i) {
      const int idx = i * 256 + tid;
      const int r = idx >> 4, c4 = idx & 15;
      *(float4*)&ldsA[r * APAD + c4 * 4] =
          *(const float4*)(Ag + (size_t)(kk + r) * DH + c4 * 4);
    }
    __syncthreads();
#pragma unroll
    for (int ks = 0; ks < KC; ks += 4) {
      const float2 af = *(const float2*)&ldsK[(wv * 16 + l16) * KPAD + ks + 2 * lh];
      v2f a; a.x = af.x; a.y = af.y;
#pragma unroll
      for (int n = 0; n < 4; ++n) {
        v2f b;
        b.x = ldsA[(ks + 2 * lh) * APAD + n * 16 + l16];
        b.y = ldsA[(ks + 2 * lh + 1) * APAD + n * 16 + l16];
        acc[n] = wmma4(a, b, acc[n]);   // A fragment reused 4x
      }
    }
    __syncthreads();
  }

  const float lam = softplus_eps(lambda_reg[0]);
#pragma unroll
  for (int n = 0; n < 4; ++n) {
    const int d = n * 16 + l16;
#pragma unroll
    for (int j = 0; j < 8; ++j) {
      const int s = ms * MSLAB + wv * 16 + j + 8 * lh;
      const size_t idx = ((size_t)bh * SS + s) * DH + d;
      if (mode == 0) outp[idx] = v[idx] - acc[n][j] - lam * alpha[idx];
      else           outp[idx] = acc[n][j];
    }
  }
}

// --- 6) ltr[bh,r,d] = sum_s lr[h,s,r] * resid[bh,s,d] ----------------------
__global__ __launch_bounds__(128) void ltr_kernel(
    const float* __restrict__ lr, const float* __restrict__ resid,
    float* __restrict__ ltr) {
  const int lane = threadIdx.x & 31, lh = lane >> 4, l16 = lane & 15;
  const int waveId = blockIdx.x * 4 + (threadIdx.x >> 5);  // B*H*4 = 128 waves
  const int bh = waveId / (DH / 16);
  const int tD = waveId % (DH / 16);
  const int h  = bh % HH;
  const float* lrb = lr    + ((size_t)h  * SS + 2 * lh) * RK + l16;
  const float* rb  = resid + ((size_t)bh * SS + 2 * lh) * DH + tD * 16 + l16;
  v8f acc = {};
  for (int ss = 0; ss < SS; ss += 4) {
    v2f a, b;
    a.x = lrb[(size_t)ss * RK];       a.y = lrb[(size_t)(ss + 1) * RK];
    b.x = rb [(size_t)ss * DH];       b.y = rb [(size_t)(ss + 1) * DH];
    acc = wmma4(a, b, acc);
  }
  const int d = tD * 16 + l16;
#pragma unroll
  for (int j = 0; j < 8; ++j) {
    const int r = j + 8 * lh;
    ltr[((size_t)bh * RK + r) * DH + d] = acc[j];
  }
}

// --- 7) alpha = clip(alpha + resid*dp + lr@ltr, +-10) ----------------------
__global__ __launch_bounds__(256) void update_kernel(
    const float* __restrict__ lr, const float* __restrict__ ltr,
    const float* __restrict__ resid, const float* __restrict__ diag_scale,
    const float* __restrict__ reg, float* __restrict__ alpha) {
  const int lane = threadIdx.x & 31, lh = lane >> 4, l16 = lane & 15;
  const int waveId = blockIdx.x * 8 + (threadIdx.x >> 5);
  const int tilesD = DH / 16, tilesS = SS / 16;
  const int bh  = waveId / (tilesS * tilesD);
  const int rem = waveId % (tilesS * tilesD);
  const int tS = rem / tilesD, tD = rem % tilesD;
  const int h  = bh % HH;
  const float* lrb  = lr  + ((size_t)h  * SS + tS * 16 + l16) * RK + 2 * lh;
  const float* ltrb = ltr + ((size_t)bh * RK + 2 * lh) * DH + tD * 16 + l16;
  v8f acc = {};
  for (int rr = 0; rr < RK; rr += 4) {
    float2 av = *(const float2*)(lrb + rr);
    v2f a, b;
    a.x = av.x; a.y = av.y;
    b.x = ltrb[(size_t)rr * DH];
    b.y = ltrb[(size_t)(rr + 1) * DH];
    acc = wmma4(a, b, acc);
  }
  // kernel diag is masked to 0 before diagonal() -> softplus(0)*scale + reg
  const float dp = 0.6931471805599453f * diag_scale[h] + reg[0];
  const int d = tD * 16 + l16;
#pragma unroll
  for (int j = 0; j < 8; ++j) {
    const int s = tS * 16 + j + 8 * lh;
    const size_t idx = ((size_t)bh * SS + s) * DH + d;
    float na = alpha[idx] + resid[idx] * dp + acc[j];
    alpha[idx] = fminf(fmaxf(na, -10.f), 10.f);
  }
}

// --- 8) XSA subtract-projection + [B,H,S,Dh] -> [B,S,H*Dh] transpose -------
__global__ void xsa_kernel(const float* __restrict__ outr,
                           const float* __restrict__ v,
                           const float* __restrict__ xsa_scale,
                           float* __restrict__ attn) {
  const int i = blockIdx.x * blockDim.x + threadIdx.x;  // row over B*H*S
  if (i >= BB * HH * SS) return;
  const int s = i % SS, h = (i / SS) % HH, b = i / (SS * HH);
  const float* op = outr + (size_t)i * DH;
  const float* vp = v    + (size_t)i * DH;
  float vn = 1e-6f, ov = 0.f;
  for (int d = 0; d < DH; ++d) { vn += vp[d] * vp[d]; ov += op[d] * vp[d]; }
  const float c = xsa_scale[0] * (ov / vn);
  float* ap = attn + ((size_t)b * SS + s) * DM + h * DH;
  for (int d = 0; d < DH; ++d) ap[d] = op[d] - c * vp[d];
}

extern "C" void kernel_launch(void* const* d_in, const int* in_sizes, int n_in,
                              void* d_out, int out_size, void* d_ws,
                              size_t ws_size, hipStream_t stream) {
  const float* x          = (const float*)d_in[0];
  const float* w_q        = (const float*)d_in[1];
  const float* w_k        = (const float*)d_in[2];
  const float* w_v        = (const float*)d_in[3];
  const float* w_o        = (const float*)d_in[4];
  const float* bandwidth  = (const float*)d_in[5];
  const float* diag_scale = (const float*)d_in[6];
  const float* pos_emb    = (const float*)d_in[7];
  const float* head_proj  = (const float*)d_in[8];
  const float* reg        = (const float*)d_in[9];
  const float* lambda_reg = (const float*)d_in[10];
  const float* xsa_scale  = (const float*)d_in[11];
  float* out = (float*)d_out;
  float* ws  = (float*)d_ws;

  float* q     = ws + OFF_Q;
  float* k     = ws + OFF_K;
  float* v     = ws + OFF_V;
  float* alpha = ws + OFF_ALPHA;
  float* resid = ws + OFF_RES;
  float* attn  = ws + OFF_ATTN;
  float* lr    = ws + OFF_LR;
  float* ltr   = ws + OFF_LTR;
  float* qsq   = ws + OFF_QSQ;
  float* ksq   = ws + OFF_KSQ;
  float* Kmat  = ws + OFF_KMAT;

  const int gemmBlocks = (BB * SS / 16) * (DM / 16) / 8;      // 2048
  const int kmatBlocks = BB * HH * (SS / 16) * (SS / 16) / 8; // 65536
  const int slabBlocks = BB * HH * (SS / MSLAB);              // 512
  const int updBlocks  = BB * HH * (SS / 16) * (DH / 16) / 8; // 2048
  const int rows       = BB * HH * SS;

  // projections
  proj_gemm_kernel<<<gemmBlocks, 256, 0, stream>>>(x, w_q, q, 0);
  proj_gemm_kernel<<<gemmBlocks, 256, 0, stream>>>(x, w_k, k, 0);
  proj_gemm_kernel<<<gemmBlocks, 256, 0, stream>>>(x, w_v, v, 0);
  // row norms + Gram/RBF matrix (diag masked), materialized once
  sumsq_kernel<<<(rows + 255) / 256, 256, 0, stream>>>(q, k, qsq, ksq);
  rbf_kernel_matrix<<<kmatBlocks, 256, 0, stream>>>(q, k, qsq, ksq, bandwidth,
                                                    Kmat);
  // low-rank preconditioner basis + alpha = 0
  lr_kernel<<<(HH * SS * RK + 255) / 256, 256, 0, stream>>>(pos_emb, head_proj,
                                                            lr);
  zero_kernel<<<(int)((QKV + 255) / 256), 256, 0, stream>>>(alpha, QKV);

  // PCG-style fixed-point iterations
  for (int it = 0; it < ITERS; ++it) {
    kmat_gemm_kernel<<<slabBlocks, 256, 0, stream>>>(Kmat, alpha, v,
                                                     lambda_reg, resid, 0);
    ltr_kernel<<<32, 128, 0, stream>>>(lr, resid, ltr);
    update_kernel<<<updBlocks, 256, 0, stream>>>(lr, ltr, resid, diag_scale,
                                                 reg, alpha);
  }

  // out = Kmat @ alpha, XSA projection removal, final output GEMM
  kmat_gemm_kernel<<<slabBlocks, 256, 0, stream>>>(Kmat, alpha, v, lambda_reg,
                                                   resid, 1);
  xsa_kernel<<<(rows + 255) / 256, 256, 0, stream>>>(resid, v, xsa_scale, attn);
  proj_gemm_kernel<<<gemmBlocks, 256, 0, stream>>>(attn, w_o, out, 1);
}